// Net_28252294873366
// MI455X (gfx1250) — compile-verified
//
#include <hip/hip_runtime.h>
#include <math.h>

// ---------------------------------------------------------------------------
// GraphSAGE 2-layer inference for gfx1250 (MI455X), f32 WMMA path.
//   h1   = x @ W1                    (WMMA f32 16x16x4, LDS-staged B)
//   hr   = relu(scatter_mean(h1) + b1)
//   h2   = hr @ W2                   (WMMA f32 16x16x4, LDS-staged B)
//   out  = log_softmax(scatter_mean(h2) + b2)
// ---------------------------------------------------------------------------

typedef __attribute__((ext_vector_type(2))) float v2f;
typedef __attribute__((ext_vector_type(8))) float v8f;

#define N_NODES 100000
#define N_EDGES 3200000
#define IN_CH   602
#define HID     16
#define OUT_CH  41

#define ROW_TILES (N_NODES / 16)   // 6250, exact
#define K_MAIN    600              // 602 = 4*150 + 2 -> main loop 600, tail 2
#define BPITCH    604              // IN_CH padded to even + 2 zero columns

// ---------------------------------------------------------------- zero fill
__global__ void zero_f32(float* __restrict__ p, size_t n) {
    size_t i = (size_t)blockIdx.x * blockDim.x + threadIdx.x;
    size_t stride = (size_t)gridDim.x * blockDim.x;
    for (; i < n; i += stride) p[i] = 0.0f;
}

// ------------------------------------------------------------- degree count
__global__ void degree_kernel(const int* __restrict__ dst, float* __restrict__ cnt) {
    int e = blockIdx.x * blockDim.x + threadIdx.x;
    if (e < N_EDGES) atomicAdd(&cnt[dst[e]], 1.0f);
}

// ----------------------------------------------------- GEMM1: h1 = x @ W1
// One wave per 16-row tile. W1 staged transposed+zero-padded in LDS so the
// inner loop is: 1x global_load_b64 (A) + 1x ds_load_b64 (B) + 1x WMMA,
// with zero predication. Tail (k=600..601) handled by one extra WMMA.
// Fragment layouts per ISA 7.12.2 (32-bit A 16x4 / B 4x16 / C 16x16):
//   lane<16 -> K={k,k+1}, lane>=16 -> K={k+2,k+3}; M/N = lane%16;
//   C vgpr r -> row r + 8*(lane>=16).
__global__ void __launch_bounds__(256) gemm1_wmma(const float* __restrict__ x,
                                                  const float* __restrict__ W1,
                                                  float* __restrict__ h1) {
    __shared__ float bT[HID * BPITCH];          // 16 x 604 f32 = 38.7 KB

    // stage: bT[n*BPITCH + k] = W1[k*16 + n]; coalesced global reads
    for (int idx = threadIdx.x; idx < IN_CH * HID; idx += 256) {
        int k = idx >> 4;
        int n = idx & 15;
        bT[n * BPITCH + k] = W1[idx];
    }
    if (threadIdx.x < 32) {                     // zero-pad columns 602..603
        int n = threadIdx.x >> 1;
        int k = IN_CH + (threadIdx.x & 1);
        bT[n * BPITCH + k] = 0.0f;
    }
    __syncthreads();

    int wave = blockIdx.x * 8 + (threadIdx.x >> 5);
    int lane = threadIdx.x & 31;
    if (wave >= ROW_TILES) return;              // wave-uniform: EXEC all-1s inside

    int row0 = wave * 16;
    int m    = lane & 15;
    int koff = (lane >> 4) * 2;
    const float* aptr = x  + (size_t)(row0 + m) * IN_CH + koff;  // 8B aligned
    const float* bptr = bT + m * BPITCH + koff;                  // 8B aligned

    v8f c = {};
#pragma unroll 2
    for (int k = 0; k < K_MAIN; k += 4) {
        v2f a = *(const v2f*)(aptr + k);
        v2f b = *(const v2f*)(bptr + k);
        c = __builtin_amdgcn_wmma_f32_16x16x4_f32(false, a, false, b,
                                                  (short)0, c, false, false);
    }
    // K tail: columns 600,601 valid -> only low half-wave carries A data;
    // B tail reads the zero-padded LDS columns for the high half-wave.
    v2f at; at.x = 0.0f; at.y = 0.0f;
    if (koff == 0) at = *(const v2f*)(aptr + K_MAIN);
    v2f bt = *(const v2f*)(bptr + K_MAIN);
    c = __builtin_amdgcn_wmma_f32_16x16x4_f32(false, at, false, bt,
                                              (short)0, c, false, false);

    int mbase = (lane >> 4) * 8;
#pragma unroll
    for (int r = 0; r < 8; ++r)
        h1[(size_t)(row0 + mbase + r) * HID + m] = c[r];
}

// ------------------------------------------- edge scatter, 16-wide features
__global__ void scatter16(const int* __restrict__ src, const int* __restrict__ dst,
                          const float* __restrict__ h1, float* __restrict__ agg) {
    size_t total = (size_t)N_EDGES * 16;
    size_t t = (size_t)blockIdx.x * blockDim.x + threadIdx.x;
    size_t stride = (size_t)gridDim.x * blockDim.x;
    for (; t < total; t += stride) {
        int e  = (int)(t >> 4);
        int ch = (int)(t & 15);
        atomicAdd(&agg[(size_t)dst[e] * HID + ch], h1[(size_t)src[e] * HID + ch]);
    }
}

// ------------------------------- mean + bias + relu (in place over agg1)
__global__ void finalize1(float* __restrict__ agg, const float* __restrict__ cnt,
                          const float* __restrict__ b1) {
    int t = blockIdx.x * blockDim.x + threadIdx.x;   // N_NODES*16 exact
    int node = t >> 4;
    int ch   = t & 15;
    float v = agg[t] / fmaxf(cnt[node], 1.0f) + b1[ch];
    agg[t] = v > 0.0f ? v : 0.0f;
}

// ----------------------------------------------------- GEMM2: h2 = hr @ W2
// 3 column tiles per 16-row tile; W2 staged transposed+padded to 48 cols in
// LDS so the K=16 loop (4 WMMAs) is predicate-free; only stores are masked.
__global__ void __launch_bounds__(256) gemm2_wmma(const float* __restrict__ hr,
                                                  const float* __restrict__ W2,
                                                  float* __restrict__ h2) {
    __shared__ float w2T[48 * HID];             // w2T[n*16 + k], 3 KB

    for (int idx = threadIdx.x; idx < 48 * HID; idx += 256) {
        int n = idx >> 4;
        int k = idx & 15;
        w2T[idx] = (n < OUT_CH) ? W2[(size_t)k * OUT_CH + n] : 0.0f;
    }
    __syncthreads();

    int wave = blockIdx.x * 8 + (threadIdx.x >> 5);
    int lane = threadIdx.x & 31;
    if (wave >= ROW_TILES * 3) return;          // wave-uniform exit

    int tm = wave / 3, tn = wave % 3;
    int row0 = tm * 16;
    int m    = lane & 15;
    int n    = tn * 16 + m;
    int koff = (lane >> 4) * 2;
    const float* aptr = hr  + (size_t)(row0 + m) * HID + koff;   // 8B aligned
    const float* bptr = w2T + n * HID + koff;                    // 8B aligned

    v8f c = {};
#pragma unroll
    for (int k = 0; k < HID; k += 4) {
        v2f a = *(const v2f*)(aptr + k);
        v2f b = *(const v2f*)(bptr + k);
        c = __builtin_amdgcn_wmma_f32_16x16x4_f32(false, a, false, b,
                                                  (short)0, c, false, false);
    }

    int mbase = (lane >> 4) * 8;
    if (n < OUT_CH) {
#pragma unroll
        for (int r = 0; r < 8; ++r)
            h2[(size_t)(row0 + mbase + r) * OUT_CH + n] = c[r];
    }
}

// ----------------------------------------- edge scatter, 41-wide (pad to 64)
__global__ void scatter41(const int* __restrict__ src, const int* __restrict__ dst,
                          const float* __restrict__ h2, float* __restrict__ out) {
    size_t total = (size_t)N_EDGES * 64;
    size_t t = (size_t)blockIdx.x * blockDim.x + threadIdx.x;
    size_t stride = (size_t)gridDim.x * blockDim.x;
    for (; t < total; t += stride) {
        int ch = (int)(t & 63);
        if (ch >= OUT_CH) continue;
        int e = (int)(t >> 6);
        atomicAdd(&out[(size_t)dst[e] * OUT_CH + ch],
                  h2[(size_t)src[e] * OUT_CH + ch]);
    }
}

// ------------------------- mean + bias + log_softmax, one wave32 per node
__global__ void logsoftmax41(float* __restrict__ out, const float* __restrict__ cnt,
                             const float* __restrict__ b2) {
    int wave = (blockIdx.x * blockDim.x + threadIdx.x) >> 5;
    int lane = threadIdx.x & 31;
    if (wave >= N_NODES) return;

    float inv = 1.0f / fmaxf(cnt[wave], 1.0f);
    float* row = out + (size_t)wave * OUT_CH;
    int c0 = lane, c1 = lane + 32;

    float v0 = row[c0] * inv + b2[c0];                       // c0 in [0,31] valid
    float v1 = (c1 < OUT_CH) ? (row[c1] * inv + b2[c1]) : -INFINITY;

    float mx = fmaxf(v0, v1);
#pragma unroll
    for (int d = 16; d >= 1; d >>= 1) mx = fmaxf(mx, __shfl_xor(mx, d, 32));

    float s = __expf(v0 - mx) + ((c1 < OUT_CH) ? __expf(v1 - mx) : 0.0f);
#pragma unroll
    for (int d = 16; d >= 1; d >>= 1) s += __shfl_xor(s, d, 32);

    float ls = __logf(s);
    row[c0] = v0 - mx - ls;
    if (c1 < OUT_CH) row[c1] = v1 - mx - ls;
}

// ---------------------------------------------------------------------------
extern "C" void kernel_launch(void* const* d_in, const int* in_sizes, int n_in,
                              void* d_out, int out_size, void* d_ws, size_t ws_size,
                              hipStream_t stream) {
    const float* x          = (const float*)d_in[0];
    const int*   edge_index = (const int*)  d_in[1];
    const float* W1         = (const float*)d_in[2];
    const float* b1         = (const float*)d_in[3];
    const float* W2         = (const float*)d_in[4];
    const float* b2         = (const float*)d_in[5];
    float*       out        = (float*)d_out;

    const int* src = edge_index;            // edge_index[0, :]
    const int* dst = edge_index + N_EDGES;  // edge_index[1, :]

    // workspace layout (floats): cnt | h1 | agg1(->hrelu) | h2   (~30 MB)
    float* ws   = (float*)d_ws;
    float* cnt  = ws;
    float* h1   = cnt  + N_NODES;
    float* agg1 = h1   + (size_t)N_NODES * HID;
    float* h2   = agg1 + (size_t)N_NODES * HID;

    // 1) zero accumulators (cnt, agg1, out)
    zero_f32<<<2048, 256, 0, stream>>>(cnt,  (size_t)N_NODES);
    zero_f32<<<2048, 256, 0, stream>>>(agg1, (size_t)N_NODES * HID);
    zero_f32<<<2048, 256, 0, stream>>>(out,  (size_t)N_NODES * OUT_CH);

    // 2) degrees (shared by both layers)
    degree_kernel<<<(N_EDGES + 255) / 256, 256, 0, stream>>>(dst, cnt);

    // 3) h1 = x @ W1  (WMMA f32 16x16x4; 6250 waves)
    gemm1_wmma<<<(ROW_TILES + 7) / 8, 256, 0, stream>>>(x, W1, h1);

    // 4) scatter-sum over edges, 16-wide
    scatter16<<<4096, 256, 0, stream>>>(src, dst, h1, agg1);

    // 5) mean + b1 + relu (in place)
    finalize1<<<(N_NODES * HID) / 256, 256, 0, stream>>>(agg1, cnt, b1);

    // 6) h2 = hrelu @ W2  (WMMA; 18750 waves)
    gemm2_wmma<<<(ROW_TILES * 3 + 7) / 8, 256, 0, stream>>>(agg1, W2, h2);

    // 7) scatter-sum over edges, 41-wide, directly into d_out
    scatter41<<<8192, 256, 0, stream>>>(src, dst, h2, out);

    // 8) mean + b2 + log_softmax, wave per node
    logsoftmax41<<<(N_NODES * 32 + 255) / 256, 256, 0, stream>>>(out, cnt, b2);
}